// VNCMDLayer_249108103948
// MI455X (gfx1250) — compile-verified
//
#include <hip/hip_runtime.h>
#include <hip/hip_bf16.h>
#include <math.h>

// ---------------------------------------------------------------------------
// VNCMD iteration for MI455X (gfx1250, wave32).
// Key optimization: opedoub = T^T T with T symmetric tridiagonal, so
// opedoub @ p = T(T p) = two 3-point stencils (O(N)), never materialize NxN.
// Each CG solve = 1 workgroup, p/q in LDS, tree reductions for dot products.
// The hyperparameter MLPs (the only real GEMMs, on the serial critical path)
// run on V_WMMA_F32_16X16X4_F32 in a single-wave kernel.
// ---------------------------------------------------------------------------

#define BB 2
#define KK 3
#define NN 2048
#define BKN (BB * KK * NN)
#define CGT 256          // threads per CG block
#define CHUNK (NN / CGT) // 8 elements per thread
#define PI2F 6.28318530717958647692f
#define CG_ITERS 30
#define CG_TOLF 1e-6f

typedef __attribute__((ext_vector_type(2))) float v2f;
typedef __attribute__((ext_vector_type(8))) float v8f;

// ------------------------------ helpers ------------------------------------

__device__ __forceinline__ float block_reduce_sum(float v, float* sred) {
  int t = threadIdx.x;
  sred[t] = v;
  __syncthreads();
  for (int s = blockDim.x >> 1; s > 0; s >>= 1) {
    if (t < s) sred[t] += sred[t + s];
    __syncthreads();
  }
  float r = sred[0];
  __syncthreads();
  return r;
}

// cumtrapz with leading zero, scaled; result per-thread in ph[CHUNK].
// y may be global or LDS (generic pointer). Uses sred[CGT] as scratch.
__device__ void cumtrapz_scan(const float* y, float dx, float scale,
                              float ph[CHUNK], float* sred) {
  int t = threadIdx.x;
  float run = 0.f;
  float half = 0.5f * dx;
#pragma unroll
  for (int j = 0; j < CHUNK; ++j) {
    int i = t * CHUNK + j;
    float term = (i == 0) ? 0.f : half * (y[i - 1] + y[i]);
    run += term;
    ph[j] = run;
  }
  sred[t] = run;
  __syncthreads();
  if (t == 0) {  // exclusive scan of 256 partials (serial: trivial cost)
    float acc = 0.f;
    for (int k = 0; k < CGT; ++k) { float tmp = sred[k]; sred[k] = acc; acc += tmp; }
  }
  __syncthreads();
  float off = sred[t];
#pragma unroll
  for (int j = 0; j < CHUNK; ++j) ph[j] = scale * (ph[j] + off);
  __syncthreads();
}

// ------------------------- K0: stats + WMMA MLP ----------------------------

__device__ __forceinline__ float red32(float v, volatile float* buf) {
  int lane = threadIdx.x;
  buf[lane] = v;
  __syncthreads();
  for (int s = 16; s > 0; s >>= 1) {
    if (lane < s) buf[lane] += buf[lane + s];
    __syncthreads();
  }
  float r = buf[0];
  __syncthreads();
  return r;
}

// A fragment: 16x4 f32. lanes 0-15: M=lane, K={k0,k0+1}; lanes 16-31: K={k0+2,k0+3}.
__device__ __forceinline__ v2f frag_a(const float* A, int rows, int kdim,
                                      int astride, int k0) {
  int lane = threadIdx.x;
  int row = lane & 15;
  int kk = k0 + ((lane >> 4) << 1);
  v2f a;
  a.x = (row < rows && kk < kdim) ? A[row * astride + kk] : 0.f;
  a.y = (row < rows && (kk + 1) < kdim) ? A[row * astride + kk + 1] : 0.f;
  return a;
}

// B fragment: 4x16 f32 of B[k][n] = W[n][k], W is (outdim, indim) row-major.
__device__ __forceinline__ v2f frag_b(const float* W, int outdim, int indim,
                                      int n0, int k0) {
  int lane = threadIdx.x;
  int col = n0 + (lane & 15);
  int kk = k0 + ((lane >> 4) << 1);
  v2f b;
  b.x = (col < outdim && kk < indim) ? W[col * indim + kk] : 0.f;
  b.y = (col < outdim && (kk + 1) < indim) ? W[col * indim + kk + 1] : 0.f;
  return b;
}

// out(rows x outdim) = act( A(rows x kdim) @ W^T + bias ), via v_wmma_f32_16x16x4_f32
__device__ void gemm_wmma(const float* A, int rows, int kdim, int astride,
                          const float* W, int outdim,
                          const float* bias, float* out, int ostride, int activ) {
  int lane = threadIdx.x;
  for (int n0 = 0; n0 < outdim; n0 += 16) {
    v8f c = {};
    for (int k0 = 0; k0 < kdim; k0 += 4) {
      v2f a = frag_a(A, rows, kdim, astride, k0);
      v2f b = frag_b(W, outdim, kdim, n0, k0);
      c = __builtin_amdgcn_wmma_f32_16x16x4_f32(false, a, false, b, (short)0, c,
                                                false, false);
    }
    // D layout: VGPR m, lanes 0-15 -> (M=m, N=lane)
    if (lane < 16 && (n0 + lane) < outdim) {
      for (int m = 0; m < rows; ++m) {
        float v = c[m] + bias[n0 + lane];
        if (activ == 1) v = fmaxf(v, 0.f);
        else if (activ == 2) v = tanhf(v);
        out[m * ostride + n0 + lane] = v;
      }
    }
    __syncthreads();
  }
}

__global__ __launch_bounds__(32) void k0_stats_mlp(
    const float* eIF, const int* mode_mask,
    const float* alpha_p, const float* beta_p,
    const float* fe_w1, const float* fe_b1, const float* fe_w2, const float* fe_b2,
    const float* pr_w1, const float* pr_b1, const float* pr_w2, const float* pr_b2,
    const float* pr_w3, const float* pr_b3,
    const float* iter_w_p, const int* iteration_p,
    float* SCAL, float* out_scalars) {
  __shared__ float buf[32];
  __shared__ float initf[BB * 4];   // (2 x 2), stride 4
  __shared__ float h1[BB * 32];
  __shared__ float cbuf[BB * 20];   // h2(16) + alpha,beta -> 18, stride 20
  __shared__ float r1[BB * 64];
  __shared__ float r2[BB * 32];
  __shared__ float rr[BB * 16];     // final 2x2, stride 16

  int lane = threadIdx.x;
  float alpha = alpha_p[0], beta = beta_p[0];
  if (lane < BB * 4) initf[lane] = 0.f;
  __syncthreads();

  // per-(b,k) mean / unbiased var of diff(eIF), masked, averaged over K
  for (int bk = 0; bk < BB * KK; ++bk) {
    const float* e = eIF + bk * NN;
    float s1 = 0.f, s2 = 0.f;
    for (int i = lane; i < NN - 1; i += 32) {
      float d = e[i + 1] - e[i];
      s1 += d;
      s2 += d * d;
    }
    float S1 = red32(s1, buf);
    float S2 = red32(s2, buf);
    if (lane == 0) {
      const float cnt = (float)(NN - 1);
      float m = S1 / cnt;
      float v = (S2 - S1 * S1 / cnt) / (cnt - 1.f);
      if (!(mode_mask[bk] > 0)) { m = 0.f; v = 0.f; }
      int b = bk / KK;
      initf[b * 4 + 0] += m * (1.f / (float)KK);
      initf[b * 4 + 1] += v * (1.f / (float)KK);
    }
    __syncthreads();
  }

  // MLP chain on V_WMMA_F32_16X16X4_F32
  gemm_wmma(initf, BB, 2, 4, fe_w1, 32, fe_b1, h1, 32, 1);
  gemm_wmma(h1, BB, 32, 32, fe_w2, 16, fe_b2, cbuf, 20, 1);
  if (lane == 0) {
    for (int b = 0; b < BB; ++b) { cbuf[b * 20 + 16] = alpha; cbuf[b * 20 + 17] = beta; }
  }
  __syncthreads();
  gemm_wmma(cbuf, BB, 18, 20, pr_w1, 64, pr_b1, r1, 64, 1);
  gemm_wmma(r1, BB, 64, 64, pr_w2, 32, pr_b2, r2, 32, 1);
  gemm_wmma(r2, BB, 32, 32, pr_w3, 2, pr_b3, rr, 16, 2);

  if (lane == 0) {
    float it = (float)iteration_p[0];
    float sig = 1.f / (1.f + expf(-iter_w_p[0] * it));
    float r00 = rr[0] * sig * 0.1f, r01 = rr[1] * sig * 0.1f;
    float r10 = rr[16] * sig * 0.1f, r11 = rr[17] * sig * 0.1f;
    float na = alpha + alpha * 0.5f * (r00 + r10);
    float nb = beta + beta * 0.5f * (r01 + r11);
    na = fminf(fmaxf(na, 1e-6f), 0.01f);
    nb = fminf(fmaxf(nb, 1e-6f), 0.1f);
    float betathr = fminf(powf(10.f, it / 36.f - 10.f), nb);
    SCAL[0] = na;
    SCAL[1] = nb;
    SCAL[2] = betathr;
    SCAL[3] = 2.f / na;        // coef for A_x/A_y
    SCAL[4] = 2.f / betathr;   // coef for S
    SCAL[5] = 1.f / na;        // inv alpha
    out_scalars[0] = na;
    out_scalars[1] = nb;
  }
}

// ------------------------- K1: projection u --------------------------------

__global__ __launch_bounds__(CGT) void k1_u(
    const float* s, const float* sum_x, const float* sum_y, const float* lam,
    const float* var_p, const float* SCAL, float* U) {
  __shared__ float sred[CGT];
  int b = blockIdx.x, t = threadIdx.x;
  int base = b * NN;
  float inva = SCAL[5];
  float dv[CHUNK];
  float ss = 0.f;
#pragma unroll
  for (int j = 0; j < CHUNK; ++j) {
    int i = t * CHUNK + j;
    float d = s[base + i] - sum_x[base + i] - sum_y[base + i] - lam[base + i] * inva;
    dv[j] = d;
    ss += d * d;
  }
  float nrm = sqrtf(block_reduce_sum(ss, sred));
  float var = var_p[0];
  float e = sqrtf((float)NN * var);
  float scale = (var == 0.f) ? 0.f : ((nrm > e) ? (e / nrm) : 1.f);
#pragma unroll
  for (int j = 0; j < CHUNK; ++j) U[base + t * CHUNK + j] = dv[j] * scale;
}

// ---------------- K2: phase scan, trig, per-mode rhs -----------------------

__global__ __launch_bounds__(CGT) void k2_rhs(
    const float* eIF, const float* xm, const float* ym, const float* s,
    const float* sum_x, const float* sum_y, const float* lam, const float* U,
    const float* fs_p, const float* SCAL,
    float* COS2, float* SIN2, float* BX, float* BY) {
  __shared__ float sred[CGT];
  int bk = blockIdx.x, t = threadIdx.x;
  int b = bk / KK;
  int mbase = bk * NN, vbase = b * NN;
  float dx = 1.f / fs_p[0];
  float inva = SCAL[5];
  float ph[CHUNK];
  cumtrapz_scan(eIF + mbase, dx, PI2F, ph, sred);
#pragma unroll
  for (int j = 0; j < CHUNK; ++j) {
    int i = t * CHUNK + j;
    float c = cosf(ph[j]), sn = sinf(ph[j]);
    float rb = s[vbase + i] - (sum_x[vbase + i] - xm[mbase + i] * c)
             - (sum_y[vbase + i] - ym[mbase + i] * sn)
             - U[vbase + i] - lam[vbase + i] * inva;
    COS2[mbase + i] = c * c;
    SIN2[mbase + i] = sn * sn;
    BX[mbase + i] = c * rb;
    BY[mbase + i] = sn * rb;
  }
}

// -------------------------- CG solver --------------------------------------
// A_reg = coef * opedoub + diag(w2) + 1e-6 I, opedoub p = T(T p), T symmetric
// tridiagonal: diag -2 (ends -1), off-diag 1.

__device__ __forceinline__ void cg_matvec(const float* sp, float* sq, float coef,
                                          const float w2[CHUNK], float Ap[CHUNK]) {
  int t = threadIdx.x;
#pragma unroll
  for (int j = 0; j < CHUNK; ++j) {
    int i = t * CHUNK + j;
    float d = (i == 0 || i == NN - 1) ? -1.f : -2.f;
    float v = d * sp[i];
    if (i > 0) v += sp[i - 1];
    if (i < NN - 1) v += sp[i + 1];
    sq[i] = v;
  }
  __syncthreads();
#pragma unroll
  for (int j = 0; j < CHUNK; ++j) {
    int i = t * CHUNK + j;
    float d = (i == 0 || i == NN - 1) ? -1.f : -2.f;
    float v = d * sq[i];
    if (i > 0) v += sq[i - 1];
    if (i < NN - 1) v += sq[i + 1];
    Ap[j] = coef * v + (w2[j] + 1e-6f) * sp[i];
  }
  __syncthreads();
}

// Two independent problem sets fused in one grid so x/y solves overlap.
__global__ __launch_bounds__(CGT) void cg_kernel(
    const float* W2a, const float* Ba, const float* X0a, float* Xa,
    const float* W2b, const float* Bb, const float* X0b, float* Xb,
    const float* SCAL, int coef_idx, int nprob) {
  __shared__ float sp[NN];
  __shared__ float sq[NN];
  __shared__ float sred[CGT];
  int pid = blockIdx.x, t = threadIdx.x;
  int set = pid / nprob, bk = pid % nprob;
  const float* W2 = set ? W2b : W2a;
  const float* Bv = set ? Bb : Ba;
  const float* X0 = set ? X0b : X0a;
  float* Xo = set ? Xb : Xa;
  int base = bk * NN;
  float coef = SCAL[coef_idx];

  float x[CHUNK], r[CHUNK], w2[CHUNK], bv[CHUNK], Ap[CHUNK];
#pragma unroll
  for (int j = 0; j < CHUNK; ++j) {
    int i = t * CHUNK + j;
    w2[j] = W2 ? W2[base + i] : 1.f;
    bv[j] = Bv[base + i];
    x[j] = X0 ? X0[base + i] : 0.f;
    sp[i] = x[j];
  }
  __syncthreads();

  cg_matvec(sp, sq, coef, w2, Ap);  // A x0
  float rs = 0.f;
#pragma unroll
  for (int j = 0; j < CHUNK; ++j) {
    r[j] = bv[j] - Ap[j];
    rs += r[j] * r[j];
  }
  float rsold = block_reduce_sum(rs, sred);
#pragma unroll
  for (int j = 0; j < CHUNK; ++j) sp[t * CHUNK + j] = r[j];  // p = r
  __syncthreads();

  for (int it = 0; it < CG_ITERS; ++it) {
    cg_matvec(sp, sq, coef, w2, Ap);
    float pap = 0.f;
#pragma unroll
    for (int j = 0; j < CHUNK; ++j) pap += sp[t * CHUNK + j] * Ap[j];
    pap = block_reduce_sum(pap, sred);
    float a = rsold / (pap + 1e-12f);
    float rs2 = 0.f;
#pragma unroll
    for (int j = 0; j < CHUNK; ++j) {
      x[j] += a * sp[t * CHUNK + j];
      r[j] -= a * Ap[j];
      rs2 += r[j] * r[j];
    }
    float rsnew = block_reduce_sum(rs2, sred);  // broadcast -> uniform branch
    if (sqrtf(rsnew) < CG_TOLF) break;          // state frozen after convergence
    float bta = rsnew / (rsold + 1e-12f);
#pragma unroll
    for (int j = 0; j < CHUNK; ++j)
      sp[t * CHUNK + j] = r[j] + bta * sp[t * CHUNK + j];
    __syncthreads();
    rsold = rsnew;
  }
#pragma unroll
  for (int j = 0; j < CHUNK; ++j) Xo[base + t * CHUNK + j] = x[j];
}

// ---------------------- K4: differ5 -> deltaIF -----------------------------

__global__ __launch_bounds__(CGT) void k4_deltaIF(
    const float* SX, const float* SY, const float* fs_p, float* DIF) {
  int bk = blockIdx.x, t = threadIdx.x;
  int base = bk * NN;
  float fs = fs_p[0];
  const float* sx = SX + base;
  const float* sy = SY + base;
#pragma unroll
  for (int j = 0; j < CHUNK; ++j) {
    int i = t * CHUNK + j;
    float xb, yb;
    if (i == 0) {
      xb = (sx[1] - sx[0]) * fs;
      yb = (sy[1] - sy[0]) * fs;
    } else if (i == NN - 1) {
      xb = (sx[NN - 1] - sx[NN - 2]) * fs;
      yb = (sy[NN - 1] - sy[NN - 2]) * fs;
    } else {
      xb = (sx[i + 1] - sx[i - 1]) * (0.5f * fs);
      yb = (sy[i + 1] - sy[i - 1]) * (0.5f * fs);
    }
    float vx = sx[i], vy = sy[i];
    DIF[base + i] = (vx * yb - vy * xb) / ((vx * vx + vy * vy + 1e-12f) * PI2F);
  }
}

// ----------------- K6: new eIF, nphase scan, cx/cy, masks ------------------

__global__ __launch_bounds__(CGT) void k6_final(
    const float* eIF, const float* xm, const float* ym, const int* mode_mask,
    const float* SX, const float* SY, const float* DSM, const float* fs_p,
    float* out_eIF, float* out_xm, float* out_ym, float* CXM, float* CYM) {
  __shared__ float ne[NN];
  __shared__ float sred[CGT];
  int bk = blockIdx.x, t = threadIdx.x;
  int base = bk * NN;
#pragma unroll
  for (int j = 0; j < CHUNK; ++j) {
    int i = t * CHUNK + j;
    ne[i] = eIF[base + i] - 0.5f * DSM[base + i];
  }
  __syncthreads();
  float ph[CHUNK];
  cumtrapz_scan(ne, 1.f / fs_p[0], PI2F, ph, sred);
  bool mk = mode_mask[bk] > 0;
#pragma unroll
  for (int j = 0; j < CHUNK; ++j) {
    int i = t * CHUNK + j;
    float sx = SX[base + i], sy = SY[base + i];
    float cx = sx * cosf(ph[j]), cy = sy * sinf(ph[j]);
    out_eIF[base + i] = mk ? ne[i] : eIF[base + i];
    out_xm[base + i] = mk ? sx : xm[base + i];
    out_ym[base + i] = mk ? sy : ym[base + i];
    CXM[base + i] = mk ? cx : 0.f;
    CYM[base + i] = mk ? cy : 0.f;
  }
}

// ------------------- K7: mode sums + lambda update -------------------------

__global__ __launch_bounds__(CGT) void k7_sums(
    const float* s, const float* lam, const float* U,
    const float* CXM, const float* CYM, const float* SCAL,
    float* out_sx, float* out_sy, float* out_lam) {
  int b = blockIdx.x, t = threadIdx.x;
  int base = b * NN;
  float na = SCAL[0];
#pragma unroll
  for (int j = 0; j < CHUNK; ++j) {
    int i = t * CHUNK + j;
    float sx = 0.f, sy = 0.f;
    for (int k = 0; k < KK; ++k) {
      sx += CXM[(b * KK + k) * NN + i];
      sy += CYM[(b * KK + k) * NN + i];
    }
    out_sx[base + i] = sx;
    out_sy[base + i] = sy;
    out_lam[base + i] = lam[base + i] + na * (U[base + i] + sx + sy - s[base + i]);
  }
}

// --------------------------------- launch ----------------------------------

extern "C" void kernel_launch(void* const* d_in, const int* in_sizes, int n_in,
                              void* d_out, int out_size, void* d_ws, size_t ws_size,
                              hipStream_t stream) {
  (void)in_sizes; (void)n_in; (void)out_size; (void)ws_size;
  const float* s     = (const float*)d_in[0];
  const float* eIF   = (const float*)d_in[1];
  const float* xm    = (const float*)d_in[2];
  const float* ym    = (const float*)d_in[3];
  const float* sum_x = (const float*)d_in[4];
  const float* sum_y = (const float*)d_in[5];
  const float* lam   = (const float*)d_in[6];
  const float* alpha = (const float*)d_in[7];
  const float* beta  = (const float*)d_in[8];
  const float* fe_w1 = (const float*)d_in[9];
  const float* fe_b1 = (const float*)d_in[10];
  const float* fe_w2 = (const float*)d_in[11];
  const float* fe_b2 = (const float*)d_in[12];
  const float* pr_w1 = (const float*)d_in[13];
  const float* pr_b1 = (const float*)d_in[14];
  const float* pr_w2 = (const float*)d_in[15];
  const float* pr_b2 = (const float*)d_in[16];
  const float* pr_w3 = (const float*)d_in[17];
  const float* pr_b3 = (const float*)d_in[18];
  const float* iter_w = (const float*)d_in[19];
  const int* mode_mask = (const int*)d_in[20];
  const float* var_p = (const float*)d_in[21];
  const float* fs_p  = (const float*)d_in[22];
  const int* iter_p  = (const int*)d_in[23];

  float* out = (float*)d_out;
  float* out_eIF = out;                  // B*K*N
  float* out_xm  = out + BKN;            // B*K*N
  float* out_ym  = out + 2 * BKN;        // B*K*N
  float* out_sx  = out + 3 * BKN;        // B*N
  float* out_sy  = out_sx + BB * NN;     // B*N
  float* out_lam = out_sy + BB * NN;     // B*N
  float* out_sc  = out_lam + BB * NN;    // 2 scalars

  float* wsf  = (float*)d_ws;
  float* SCAL = wsf;                     // 8 scalars
  float* U    = wsf + 8;                 // B*N
  float* COS2 = U + BB * NN;
  float* SIN2 = COS2 + BKN;
  float* BX   = SIN2 + BKN;
  float* BY   = BX + BKN;
  float* SX   = BY + BKN;                // solved_x
  float* SY   = SX + BKN;                // solved_y
  float* DIF  = SY + BKN;                // deltaIF
  float* DSM  = DIF + BKN;               // d_smooth
  float* CXM  = DSM + BKN;
  float* CYM  = CXM + BKN;

  // K0: stats + WMMA MLP -> new_alpha/new_beta/betathr + coefficients
  k0_stats_mlp<<<1, 32, 0, stream>>>(eIF, mode_mask, alpha, beta,
                                     fe_w1, fe_b1, fe_w2, fe_b2,
                                     pr_w1, pr_b1, pr_w2, pr_b2, pr_w3, pr_b3,
                                     iter_w, iter_p, SCAL, out_sc);
  // K1: projection u
  k1_u<<<BB, CGT, 0, stream>>>(s, sum_x, sum_y, lam, var_p, SCAL, U);
  // K2: phase scan + trig + rhs
  k2_rhs<<<BB * KK, CGT, 0, stream>>>(eIF, xm, ym, s, sum_x, sum_y, lam, U,
                                      fs_p, SCAL, COS2, SIN2, BX, BY);
  // K3: fused x & y CG solves (12 blocks, coef = 2/new_alpha)
  cg_kernel<<<2 * BB * KK, CGT, 0, stream>>>(COS2, BX, xm, SX,
                                             SIN2, BY, ym, SY,
                                             SCAL, 3, BB * KK);
  // K4: deltaIF
  k4_deltaIF<<<BB * KK, CGT, 0, stream>>>(SX, SY, fs_p, DIF);
  // K5: smoothing solve (coef = 2/betathr, w2 = 1, x0 = 0)
  cg_kernel<<<BB * KK, CGT, 0, stream>>>((const float*)nullptr, DIF,
                                         (const float*)nullptr, DSM,
                                         (const float*)nullptr, DIF,
                                         (const float*)nullptr, DSM,
                                         SCAL, 4, BB * KK);
  // K6: finalize per-mode outputs
  k6_final<<<BB * KK, CGT, 0, stream>>>(eIF, xm, ym, mode_mask, SX, SY, DSM,
                                        fs_p, out_eIF, out_xm, out_ym, CXM, CYM);
  // K7: sums over modes + lambda
  k7_sums<<<BB, CGT, 0, stream>>>(s, lam, U, CXM, CYM, SCAL,
                                  out_sx, out_sy, out_lam);
}